// Attention_82789789597885
// MI455X (gfx1250) — compile-verified
//
#include <hip/hip_runtime.h>

#define HEADS 8
#define C64 64
#define NPIX 65536            // 256*256
#define INNER 512
#define EPS 1e-12f

typedef __attribute__((ext_vector_type(2))) float v2f;
typedef __attribute__((ext_vector_type(8))) float v8f;
typedef __attribute__((ext_vector_type(4))) int   v4i;

#define KCHUNKS 64
#define CK (NPIX / KCHUNKS)   // 1024 k-elements per block
#define KC 64                 // LDS stage depth (double buffered)
#define KCPAD 68              // stride mod 64 == 4 -> conflict-free b64 reads
#define NSTAGE (CK / KC)      // 16 stages per block

#if defined(__gfx1250__) && __has_builtin(__builtin_amdgcn_global_load_async_to_lds_b128) && __has_builtin(__builtin_amdgcn_s_wait_asynccnt)
#define HAVE_ASYNC_LDS 1
#endif

#ifdef HAVE_ASYNC_LDS
__device__ __forceinline__ void async_copy16(const float* g, float* l) {
    // GLOBAL_LOAD_ASYNC_TO_LDS_B128: 16B per lane, LDS written by async engine,
    // tracked with ASYNCcnt (no VGPR round trip).
    __builtin_amdgcn_global_load_async_to_lds_b128(
        (__attribute__((address_space(1))) v4i*)g,
        (__attribute__((address_space(3))) v4i*)l, 0, 0);
}
#endif

// ---------------------------------------------------------------------------
// zero the Gram accumulator region of the workspace (ws is poisoned by harness)
__global__ void zero_ws_kernel(float* g, int n) {
    int i = blockIdx.x * blockDim.x + threadIdx.x;
    if (i < n) g[i] = 0.f;
}

// ---------------------------------------------------------------------------
// Kernel 1: G[b] = X X^T  (X = x_in[b] as [64, 65536], row-major over pixels)
// 16 waves per block, each wave owns one 16x16 tile of the 64x64 Gram matrix.
// f32 WMMA 16x16x4; k staged through double-buffered LDS filled by
// GLOBAL_LOAD_ASYNC_TO_LDS_B128; partials combined with atomics.
__global__ __launch_bounds__(512) void gram_kernel(const float* __restrict__ x,
                                                   float* __restrict__ G) {
    __shared__ float sX[2][C64][KCPAD];
    const int b     = blockIdx.x / KCHUNKS;
    const int chunk = blockIdx.x % KCHUNKS;
    const float* xb = x + (size_t)b * C64 * NPIX;
    float* Gb       = G + b * C64 * C64;

    const int t    = threadIdx.x;
    const int wave = t >> 5;          // 0..15
    const int lane = t & 31;
    const int ti   = wave >> 2;       // c1 tile
    const int tj   = wave & 3;        // c2 tile
    const int m    = lane & 15;       // M for A, N for B
    const int ks0  = (lane < 16) ? 0 : 2;

    // per-thread stage-load coordinates: 64 rows x 16 float4 cols, 2 per thread
    const int lc = t >> 4;            // row 0..31 (+32 on second pass)
    const int lf = (t & 15) * 4;      // col 0..60

    v8f acc = {};
    const int kbase = chunk * CK;

#ifdef HAVE_ASYNC_LDS
    // prologue: issue async fills for stage 0
    for (int pp = 0; pp < 2; ++pp)
        async_copy16(xb + (size_t)(lc + pp * 32) * NPIX + kbase + lf,
                     &sX[0][lc + pp * 32][lf]);
    for (int it = 0; it < NSTAGE; ++it) {
        const int cur = it & 1;
        if (it + 1 < NSTAGE) {
            const int kn = kbase + (it + 1) * KC;
            for (int pp = 0; pp < 2; ++pp)
                async_copy16(xb + (size_t)(lc + pp * 32) * NPIX + kn + lf,
                             &sX[cur ^ 1][lc + pp * 32][lf]);
            __builtin_amdgcn_s_wait_asynccnt(2);   // stage `cur` complete
        } else {
            __builtin_amdgcn_s_wait_asynccnt(0);
        }
        __syncthreads();                           // publish stage to all waves
        for (int kk = 0; kk < KC; kk += 4) {
            v2f a  = *(const v2f*)&sX[cur][ti * 16 + m][kk + ks0];
            v2f bv = *(const v2f*)&sX[cur][tj * 16 + m][kk + ks0];
            acc = __builtin_amdgcn_wmma_f32_16x16x4_f32(
                false, a, false, bv, (short)0, acc, false, false);
        }
        __syncthreads();                           // done reading before refill
    }
#else
    for (int it = 0; it < NSTAGE; ++it) {
        const int cur = it & 1;
        const int k0 = kbase + it * KC;
        __syncthreads();
        for (int pp = 0; pp < 2; ++pp) {
            int c = lc + pp * 32;
            float4 v = *(const float4*)(xb + (size_t)c * NPIX + k0 + lf);
            sX[cur][c][lf + 0] = v.x; sX[cur][c][lf + 1] = v.y;
            sX[cur][c][lf + 2] = v.z; sX[cur][c][lf + 3] = v.w;
        }
        __syncthreads();
        for (int kk = 0; kk < KC; kk += 4) {
            v2f a  = *(const v2f*)&sX[cur][ti * 16 + m][kk + ks0];
            v2f bv = *(const v2f*)&sX[cur][tj * 16 + m][kk + ks0];
            acc = __builtin_amdgcn_wmma_f32_16x16x4_f32(
                false, a, false, bv, (short)0, acc, false, false);
        }
    }
#endif

    const int hi = (lane < 16) ? 0 : 8;
    for (int r = 0; r < 8; ++r)
        atomicAdd(&Gb[(ti * 16 + r + hi) * C64 + tj * 16 + m], acc[r]);
}

// ---------------------------------------------------------------------------
// Kernel 2: per batch, derive logits from G, softmax, fold into M = Wv' @ Wp.
// Tiny (a few MFLOP) — one 256-thread block per batch, scalar math.
__global__ __launch_bounds__(256) void attn_fold_kernel(
    const float* __restrict__ G,  const float* __restrict__ Wq,
    const float* __restrict__ Wk, const float* __restrict__ Wv,
    const float* __restrict__ Wp, const float* __restrict__ rescale,
    float* __restrict__ M) {
    __shared__ float sG[C64][C64];
    __shared__ float sT[C64][C64 + 1];   // Tq = G @ Wq_h, later reused for V'
    __shared__ float sA[C64][C64 + 1];   // Tk, then logits, then softmax A
    __shared__ float s_nq[C64], s_nk[C64];

    const int b = blockIdx.x;
    const int t = threadIdx.x;
    const float* Gb = G + b * 4096;

    float regM[16];
    for (int j = 0; j < 16; ++j) regM[j] = 0.f;
    for (int i = t; i < 4096; i += 256) sG[i >> 6][i & 63] = Gb[i];
    __syncthreads();

    for (int h = 0; h < HEADS; ++h) {
        const int hb = h * 64;
        // Tq, Tk
        for (int i = t; i < 4096; i += 256) {
            int r = i >> 6, e = i & 63;
            float aq = 0.f, ak = 0.f;
            for (int c2 = 0; c2 < 64; ++c2) {
                float g = sG[r][c2];
                aq += g * Wq[c2 * INNER + hb + e];
                ak += g * Wk[c2 * INNER + hb + e];
            }
            sT[r][e] = aq;   // Tq
            sA[r][e] = ak;   // Tk
        }
        __syncthreads();
        // row norms of Q and K over the n axis (sqrt of bilinear forms)
        if (t < 64) {
            float sq = 0.f, sk = 0.f;
            for (int c1 = 0; c1 < 64; ++c1) {
                sq += Wq[c1 * INNER + hb + t] * sT[c1][t];
                sk += Wk[c1 * INNER + hb + t] * sA[c1][t];
            }
            s_nq[t] = fmaxf(sqrtf(fmaxf(sq, 0.f)), EPS);
            s_nk[t] = fmaxf(sqrtf(fmaxf(sk, 0.f)), EPS);
        }
        __syncthreads();
        // logits[d][e] = (Wk_h^T Tq)[d][e] * rescale / (nk[d]*nq[e])
        const float rs = rescale[h];
        for (int i = t; i < 4096; i += 256) {
            int d = i >> 6, e = i & 63;
            float s = 0.f;
            for (int c1 = 0; c1 < 64; ++c1)
                s += Wk[c1 * INNER + hb + d] * sT[c1][e];
            sA[d][e] = s * rs / (s_nk[d] * s_nq[e]);
        }
        __syncthreads();
        // softmax over e, one thread per row d
        if (t < 64) {
            float mx = -INFINITY;
            for (int e = 0; e < 64; ++e) mx = fmaxf(mx, sA[t][e]);
            float sum = 0.f;
            for (int e = 0; e < 64; ++e) {
                float ex = __expf(sA[t][e] - mx);
                sA[t][e] = ex; sum += ex;
            }
            float inv = 1.f / sum;
            for (int e = 0; e < 64; ++e) sA[t][e] *= inv;
        }
        __syncthreads();
        // V'[c1][d] = sum_e Wv[c1,he] A[d][e]   (reuse sT)
        for (int i = t; i < 4096; i += 256) {
            int c1 = i >> 6, d = i & 63;
            float s = 0.f;
            for (int e = 0; e < 64; ++e)
                s += Wv[c1 * INNER + hb + e] * sA[d][e];
            sT[c1][d] = s;
        }
        __syncthreads();
        // M[c1][c2] += sum_d V'[c1][d] * Wp[hd][c2]  (register accumulators)
        for (int i = t, j = 0; i < 4096; i += 256, ++j) {
            int c1 = i >> 6, c2 = i & 63;
            float s = 0.f;
            for (int d = 0; d < 64; ++d)
                s += sT[c1][d] * Wp[(hb + d) * C64 + c2];
            regM[j] += s;
        }
        __syncthreads();
    }
    float* Mb = M + b * 4096;
    for (int i = t, j = 0; i < 4096; i += 256, ++j) Mb[i] = regM[j];
}

// ---------------------------------------------------------------------------
// Kernel 3: out[b,c2,p] = (x[b,:,p]^T @ M[b])[c2] + bp[c2] + dwconv3x3(x)[c2,p]
// 64-pixel strip per block (within one image row), 8 waves -> 16 WMMA tiles.
__global__ __launch_bounds__(256) void final_kernel(
    const float* __restrict__ x,  const float* __restrict__ M,
    const float* __restrict__ bp, const float* __restrict__ posk,
    float* __restrict__ out) {
    __shared__ float sA[C64][68];   // [p][c1]  (x tile, transposed)
    __shared__ float sB[C64][68];   // [c2][c1] (M transposed)
    __shared__ float sO[C64][68];   // [c2][p]

    const int b  = blockIdx.y;
    const int p0 = blockIdx.x * 64;
    const float* xb = x + (size_t)b * C64 * NPIX;
    const int t = threadIdx.x;

    // load x tile (coalesced), store transposed for A-matrix access
    for (int pp = 0; pp < 4; ++pp) {
        int c = (t >> 4) + pp * 16;
        int f = (t & 15) * 4;
        float4 v = *(const float4*)(xb + (size_t)c * NPIX + p0 + f);
        sA[f + 0][c] = v.x; sA[f + 1][c] = v.y;
        sA[f + 2][c] = v.z; sA[f + 3][c] = v.w;
    }
    const float* Mb = M + b * 4096;
    for (int i = t; i < 4096; i += 256) {
        int c1 = i >> 6, c2 = i & 63;
        sB[c2][c1] = Mb[i];
    }
    __syncthreads();

    const int wave = t >> 5;   // 0..7
    const int lane = t & 31;
    const int m    = lane & 15;
    const int ks0  = (lane < 16) ? 0 : 2;
    const int hi   = (lane < 16) ? 0 : 8;
    for (int tt = wave * 2; tt < wave * 2 + 2; ++tt) {
        int pi = tt >> 2, cj = tt & 3;
        v8f acc = {};
        for (int kk = 0; kk < 64; kk += 4) {
            v2f a  = *(const v2f*)&sA[pi * 16 + m][kk + ks0];
            v2f bv = *(const v2f*)&sB[cj * 16 + m][kk + ks0];
            acc = __builtin_amdgcn_wmma_f32_16x16x4_f32(
                false, a, false, bv, (short)0, acc, false, false);
        }
        for (int r = 0; r < 8; ++r)
            sO[cj * 16 + m][pi * 16 + r + hi] = acc[r];
    }
    __syncthreads();

    // epilogue: bias + depthwise 3x3 conv (SAME), coalesced stores
    const int y0 = p0 >> 8;       // strip lies in one image row
    const int x0 = p0 & 255;
    const int c2 = t >> 2;
    const int pxbase = (t & 3) * 16;
    const float bias = bp[c2];
    const float* kp = posk + c2 * 9;
    float k00 = kp[0], k01 = kp[1], k02 = kp[2];
    float k10 = kp[3], k11 = kp[4], k12 = kp[5];
    float k20 = kp[6], k21 = kp[7], k22 = kp[8];
    const float* xc = xb + (size_t)c2 * NPIX;
    float* oc = out + ((size_t)b * C64 + c2) * NPIX;
    for (int i = 0; i < 16; ++i) {
        int px = pxbase + i;
        int xx = x0 + px;
        float conv = 0.f;
        for (int dy = -1; dy <= 1; ++dy) {
            int yy = y0 + dy;
            if (yy < 0 || yy > 255) continue;
            const float* row = xc + yy * 256;
            float kk0 = (dy == -1) ? k00 : ((dy == 0) ? k10 : k20);
            float kk1 = (dy == -1) ? k01 : ((dy == 0) ? k11 : k21);
            float kk2 = (dy == -1) ? k02 : ((dy == 0) ? k12 : k22);
            if (xx > 0)   conv += row[xx - 1] * kk0;
            conv += row[xx] * kk1;
            if (xx < 255) conv += row[xx + 1] * kk2;
        }
        oc[p0 + px] = sO[c2][px] + bias + conv;
    }
}

// ---------------------------------------------------------------------------
extern "C" void kernel_launch(void* const* d_in, const int* in_sizes, int n_in,
                              void* d_out, int out_size, void* d_ws, size_t ws_size,
                              hipStream_t stream) {
    const float* x_in    = (const float*)d_in[0];
    const float* Wq      = (const float*)d_in[1];
    const float* Wk      = (const float*)d_in[2];
    const float* Wv      = (const float*)d_in[3];
    const float* rescale = (const float*)d_in[4];
    const float* Wp      = (const float*)d_in[5];
    const float* bp      = (const float*)d_in[6];
    const float* pos_k   = (const float*)d_in[7];
    float* out = (float*)d_out;

    float* G = (float*)d_ws;            // 2 * 64 * 64 floats
    float* M = G + 2 * C64 * C64;       // 2 * 64 * 64 floats

    zero_ws_kernel<<<(2 * C64 * C64 + 255) / 256, 256, 0, stream>>>(G, 2 * C64 * C64);
    gram_kernel<<<2 * KCHUNKS, 512, 0, stream>>>(x_in, G);
    attn_fold_kernel<<<2, 256, 0, stream>>>(G, Wq, Wk, Wv, Wp, rescale, M);
    final_kernel<<<dim3(NPIX / 64, 2), 256, 0, stream>>>(x_in, M, bp, pos_k, out);

    (void)in_sizes; (void)n_in; (void)out_size; (void)ws_size;
}